// GCNProteinsModel_36867999269112
// MI455X (gfx1250) — compile-verified
//
#include <hip/hip_runtime.h>
#include <hip/hip_bf16.h>

// ---------------------------------------------------------------------------
// GCN (2x GCNConv + mean pool) for MI455X / gfx1250, wave32.
// - GEMM1 (100000x256 @ 256x16) uses V_WMMA_F32_16X16X4_F32 (exact fp32),
//   with non-temporal x loads (x streamed once; keep L2 for graph data).
// - Edge aggregation: float4-vectorized gather + global_atomic_add_f32.
//   h1/agg1 (6.4MB each) are L2-resident (192MB L2), so reducing issued
//   memory ops per edge (not just HBM bytes) is what matters.
// ---------------------------------------------------------------------------

typedef __attribute__((ext_vector_type(2))) float v2f;
typedef __attribute__((ext_vector_type(8))) float v8f;

namespace {
constexpr long long NN  = 100000;   // nodes
constexpr long long NE  = 3200000;  // edges
constexpr int       NG  = 512;      // graphs
constexpr int       FIN = 256;
constexpr int       HID = 16;       // log2 = 4
constexpr int       NC  = 8;        // log2 = 3
}

// ---------------- generic fill ----------------
__global__ void fill_kernel(float* __restrict__ p, float v, long long n) {
  long long i = (long long)blockIdx.x * blockDim.x + threadIdx.x;
  if (i < n) p[i] = v;
}

// ---------------- degree (self-loop pre-counted via init=1.0) --------------
__global__ void degree_kernel(const long long* __restrict__ dst,
                              float* __restrict__ deg, long long E) {
  long long e = (long long)blockIdx.x * blockDim.x + threadIdx.x;
  if (e < E) atomicAdd(&deg[dst[e]], 1.0f);
}

__global__ void dinv_kernel(float* __restrict__ deg, long long n) {
  long long i = (long long)blockIdx.x * blockDim.x + threadIdx.x;
  if (i < n) deg[i] = rsqrtf(fmaxf(deg[i], 1.0f));
}

// ---------------- GEMM1: h1[NN,16] = x[NN,256] @ W1[256,16] via WMMA -------
// One wave per 16-row node tile; NN/16 = 6250 tiles exactly.
// f32 16x16x4 fragment layouts (ISA 7.12.2):
//   A (16x4):  lanes 0-15 -> {K0,K1}, lanes 16-31 -> {K2,K3}, M = lane&15
//   B (4x16):  v0/v1 hold rows {K0,K1} (lanes 0-15) / {K2,K3} (lanes 16-31)
//   C/D: vgpr r, lanes 0-15 -> M=r, lanes 16-31 -> M=r+8, N = lane&15
__global__ void gemm1_wmma_kernel(const float* __restrict__ x,
                                  const float* __restrict__ W1,
                                  float* __restrict__ h1) {
  const int wavesPerBlock = blockDim.x >> 5;
  const int wave = blockIdx.x * wavesPerBlock + (threadIdx.x >> 5);
  const int lane = threadIdx.x & 31;
  const int col  = lane & 15;       // M for A, N for B/D
  const int hi   = lane >> 4;       // selects K pair {0,1} vs {2,3}

  const long long row = (long long)wave * 16 + col;
  const float* __restrict__ xr = x + row * FIN + 2 * hi;

  v8f c = {};
#pragma unroll 4
  for (int k = 0; k < FIN; k += 4) {
    // x is streamed exactly once -> non-temporal, keep L2 for graph data.
    v2f a = __builtin_nontemporal_load((const v2f*)(xr + k));
    v2f b;
    b.x = W1[(k + 2 * hi    ) * HID + col];
    b.y = W1[(k + 2 * hi + 1) * HID + col];
    c = __builtin_amdgcn_wmma_f32_16x16x4_f32(
        /*neg_a=*/false, a, /*neg_b=*/false, b,
        /*c_mod=*/(short)0, c, /*reuse_a=*/false, /*reuse_b=*/false);
  }

  const long long base = (long long)wave * 16;
#pragma unroll
  for (int r = 0; r < 8; ++r)
    h1[(base + r + 8 * hi) * HID + col] = c[r];
}

// ---------------- edge scatter: out[d] += h[s] * dinv[s]*dinv[d] -----------
// 4 features per thread: one b128 gather + 4 clause'd f32 atomics.
// LOGF = log2(features); quads-per-edge = 1 << (LOGF-2).
template <int LOGF>
__global__ void scatter_edges_vec4_kernel(const long long* __restrict__ src,
                                          const long long* __restrict__ dst,
                                          const float* __restrict__ dinv,
                                          const float* __restrict__ h,
                                          float* __restrict__ out,
                                          long long totalQuads) {
  long long gid = (long long)blockIdx.x * blockDim.x + threadIdx.x;
  if (gid >= totalQuads) return;
  constexpr int LOGQ = LOGF - 2;
  long long e = gid >> LOGQ;
  int q = (int)(gid & ((1 << LOGQ) - 1));
  long long s = src[e];
  long long d = dst[e];
  float norm = dinv[s] * dinv[d];
  const float4 hv = *(const float4*)(h + (s << LOGF) + q * 4);
  float* o = out + (d << LOGF) + q * 4;
  atomicAdd(o + 0, hv.x * norm);
  atomicAdd(o + 1, hv.y * norm);
  atomicAdd(o + 2, hv.z * norm);
  atomicAdd(o + 3, hv.w * norm);
}

// self-loop pass: runs after edge scatter (stream ordered), one thread per
// float4 -> plain read-modify-write, no atomic needed.
template <int LOGF>
__global__ void self_loop_vec4_kernel(const float* __restrict__ dinv,
                                      const float* __restrict__ h,
                                      float* __restrict__ out,
                                      long long totalQuads) {
  long long gid = (long long)blockIdx.x * blockDim.x + threadIdx.x;
  if (gid >= totalQuads) return;
  long long node = gid >> (LOGF - 2);
  float dv = dinv[node];
  float n2 = dv * dv;
  float4 hv = *(const float4*)(h + gid * 4);
  float4* op = (float4*)(out + gid * 4);
  float4 ov = *op;
  ov.x += hv.x * n2;
  ov.y += hv.y * n2;
  ov.z += hv.z * n2;
  ov.w += hv.w * n2;
  *op = ov;
}

// ---------------- bias + relu (layer 1 epilogue) ---------------------------
__global__ void bias_relu_kernel(float* __restrict__ a, const float* __restrict__ b,
                                 long long total, int mask) {
  long long gid = (long long)blockIdx.x * blockDim.x + threadIdx.x;
  if (gid >= total) return;
  a[gid] = fmaxf(a[gid] + b[gid & mask], 0.0f);
}

// ---------------- GEMM2: h2[NN,8] = agg1[NN,16] @ W2[16,8] (trivial) -------
__global__ void gemm2_kernel(const float* __restrict__ agg1,
                             const float* __restrict__ W2,
                             float* __restrict__ h2, long long total) {
  long long gid = (long long)blockIdx.x * blockDim.x + threadIdx.x;
  if (gid >= total) return;
  long long node = gid >> 3;
  int cls = (int)(gid & 7);
  const float* __restrict__ row = agg1 + node * HID;
  float s = 0.0f;
#pragma unroll
  for (int k = 0; k < HID; ++k) s = fmaf(row[k], W2[k * NC + cls], s);
  h2[gid] = s;
}

// ---------------- mean pool over graphs ------------------------------------
__global__ void pool_acc_kernel(const float* __restrict__ h2,
                                const long long* __restrict__ batch,
                                float* __restrict__ sums, float* __restrict__ cnt,
                                long long total) {
  long long gid = (long long)blockIdx.x * blockDim.x + threadIdx.x;
  if (gid >= total) return;
  long long node = gid >> 3;
  int f = (int)(gid & 7);
  long long g = batch[node];
  atomicAdd(&sums[g * NC + f], h2[gid]);
  if (f == 0) atomicAdd(&cnt[g], 1.0f);
}

// b2 is linear through the mean, so it is added here (once per graph cell).
// Empty graphs produce exactly 0 as in the reference.
__global__ void pool_final_kernel(const float* __restrict__ sums,
                                  const float* __restrict__ cnt,
                                  const float* __restrict__ b2,
                                  float* __restrict__ out, long long total) {
  long long gid = (long long)blockIdx.x * blockDim.x + threadIdx.x;
  if (gid >= total) return;
  long long g = gid >> 3;
  int f = (int)(gid & 7);
  float c = cnt[g];
  out[gid] = (c > 0.0f) ? (sums[gid] / c + b2[f]) : 0.0f;
}

// ---------------------------------------------------------------------------
static inline unsigned int nblk(long long total, int bs) {
  return (unsigned int)((total + bs - 1) / bs);
}

extern "C" void kernel_launch(void* const* d_in, const int* in_sizes, int n_in,
                              void* d_out, int out_size, void* d_ws, size_t ws_size,
                              hipStream_t stream) {
  (void)in_sizes; (void)n_in; (void)out_size; (void)ws_size;

  const float*      x     = (const float*)d_in[0];
  const long long*  ei    = (const long long*)d_in[1];      // (2, NE) int64
  const long long*  batch = (const long long*)d_in[2];      // (NN,)  int64
  /* d_in[3] = num_graphs scalar (512), known at compile time */
  const float*      W1    = (const float*)d_in[4];          // (256,16)
  const float*      b1    = (const float*)d_in[5];          // (16,)
  const float*      W2    = (const float*)d_in[6];          // (16,8)
  const float*      b2    = (const float*)d_in[7];          // (8,)
  float*            out   = (float*)d_out;                  // (512,8)

  const long long* src = ei;        // edge_index[0]
  const long long* dst = ei + NE;   // edge_index[1]

  // Workspace carve-up (~20 MB total)
  char* w = (char*)d_ws;
  float* dinv = (float*)w; w += NN * 4;            // degree -> rsqrt in place
  float* h1   = (float*)w; w += NN * HID * 4;      // x @ W1
  float* agg1 = (float*)w; w += NN * HID * 4;      // aggregated layer 1
  float* h2   = (float*)w; w += NN * NC * 4;       // agg1 @ W2
  float* agg2 = (float*)w; w += NN * NC * 4;       // aggregated layer 2
  float* sums = (float*)w; w += (long long)NG * NC * 4;
  float* cnt  = (float*)w; w += (long long)NG * 4;

  const int BS = 256;

  // 1) degrees (init 1.0 = self-loop) -> dinv = rsqrt(max(deg,1))
  fill_kernel<<<nblk(NN, BS), BS, 0, stream>>>(dinv, 1.0f, NN);
  degree_kernel<<<nblk(NE, BS), BS, 0, stream>>>(dst, dinv, NE);
  dinv_kernel<<<nblk(NN, BS), BS, 0, stream>>>(dinv, NN);

  // 2) GEMM1 via WMMA: 6250 tiles of 16 rows; 10 waves (320 threads) / block.
  gemm1_wmma_kernel<<<625, 320, 0, stream>>>(x, W1, h1);

  // 3) layer-1 aggregation (float4-vectorized: 4 quads/edge)
  fill_kernel<<<nblk(NN * HID, BS), BS, 0, stream>>>(agg1, 0.0f, NN * HID);
  scatter_edges_vec4_kernel<4><<<nblk(NE * 4, BS), BS, 0, stream>>>(src, dst, dinv, h1, agg1, NE * 4);
  self_loop_vec4_kernel<4><<<nblk(NN * 4, BS), BS, 0, stream>>>(dinv, h1, agg1, NN * 4);
  bias_relu_kernel<<<nblk(NN * HID, BS), BS, 0, stream>>>(agg1, b1, NN * HID, HID - 1);

  // 4) GEMM2 (tiny) + layer-2 aggregation (2 quads/edge)
  gemm2_kernel<<<nblk(NN * NC, BS), BS, 0, stream>>>(agg1, W2, h2, NN * NC);
  fill_kernel<<<nblk(NN * NC, BS), BS, 0, stream>>>(agg2, 0.0f, NN * NC);
  scatter_edges_vec4_kernel<3><<<nblk(NE * 2, BS), BS, 0, stream>>>(src, dst, dinv, h2, agg2, NE * 2);
  self_loop_vec4_kernel<3><<<nblk(NN * 2, BS), BS, 0, stream>>>(dinv, h2, agg2, NN * 2);

  // 5) global mean pool (+ b2 folded into finalize)
  fill_kernel<<<nblk((long long)NG * NC, BS), BS, 0, stream>>>(sums, 0.0f, (long long)NG * NC);
  fill_kernel<<<nblk((long long)NG, BS), BS, 0, stream>>>(cnt, 0.0f, (long long)NG);
  pool_acc_kernel<<<nblk(NN * NC, BS), BS, 0, stream>>>(agg2, batch, sums, cnt, NN * NC);
  pool_final_kernel<<<nblk((long long)NG * NC, BS), BS, 0, stream>>>(sums, cnt, b2, out, (long long)NG * NC);
}